// GCN1D_Encoder_43379169689748
// MI455X (gfx1250) — compile-verified
//
#include <hip/hip_runtime.h>
#include <cstddef>

typedef __attribute__((ext_vector_type(2))) float v2f;
typedef __attribute__((ext_vector_type(8))) float v8f;

static constexpr int N_NODES = 10000;
static constexpr int N_EDGES = 80000;
static constexpr int MAXF    = 2304;      // largest activation row (96ch * 24t)
static constexpr int PRED    = 24;
static constexpr int LINF    = 96 * 22;   // 2112 (time-major == packed layer-5 output)

// ---------------------------------------------------------------- utilities
__global__ void zero_f(float* __restrict__ p, int n) {
  for (int i = blockIdx.x * blockDim.x + threadIdx.x; i < n; i += gridDim.x * blockDim.x)
    p[i] = 0.0f;
}

__global__ void deg_acc(float* __restrict__ deg, const int* __restrict__ src, int E) {
  int i = blockIdx.x * blockDim.x + threadIdx.x;
  if (i < E) atomicAdd(&deg[src[i]], 1.0f);
}

__global__ void dinv_k(float* __restrict__ deg, int N) {
  int i = blockIdx.x * blockDim.x + threadIdx.x;
  if (i < N) { float d = deg[i]; deg[i] = (d > 0.0f) ? rsqrtf(d) : 0.0f; }
}

__global__ void edgew(float* __restrict__ w, const float* __restrict__ dinv,
                      const int* __restrict__ src, const int* __restrict__ dst, int E) {
  int i = blockIdx.x * blockDim.x + threadIdx.x;
  if (i < E) w[i] = -(dinv[src[i]] * dinv[dst[i]]);
}

__global__ void axpby(float* __restrict__ out, const float* __restrict__ X,
                      const float* __restrict__ Y, float a, float b, size_t n) {
  size_t i0 = (size_t)blockIdx.x * blockDim.x + threadIdx.x;
  size_t st = (size_t)gridDim.x * blockDim.x;
  if (Y) { for (size_t i = i0; i < n; i += st) out[i] = a * X[i] + b * Y[i]; }
  else   { for (size_t i = i0; i < n; i += st) out[i] = a * X[i]; }
}

// out[dst[e], :] += scale * w[e] * h[src[e], :]  — one wave per edge, lanes over features
__global__ void scatter_add(float* __restrict__ out, const float* __restrict__ h,
                            const int* __restrict__ src, const int* __restrict__ dst,
                            const float* __restrict__ w, float scale, int E, int F) {
  int gtid = blockIdx.x * blockDim.x + threadIdx.x;
  int wid  = gtid >> 5;
  int lane = threadIdx.x & 31;
  int nw   = (gridDim.x * blockDim.x) >> 5;
  for (int e = wid; e < E; e += nw) {
    int s = src[e], d = dst[e];
    if (e + nw < E)  // prefetch next source row (global_prefetch_b8 path)
      __builtin_prefetch(&h[(size_t)src[e + nw] * F], 0, 1);
    float wt = w[e] * scale;
    const float* hs = h + (size_t)s * F;
    float* od = out + (size_t)d * F;
    for (int f = lane; f < F; f += 32) atomicAdd(&od[f], wt * hs[f]);
  }
}

// ------------------------------------------------------------ pack kernels
// x[N][5][24] (channel-major) -> xp[N][24][8] (time-major, ch 5..7 zero)
__global__ void pack_x(float* __restrict__ xp, const float* __restrict__ x) {
  const int total = N_NODES * 24 * 8;
  for (int i = blockIdx.x * blockDim.x + threadIdx.x; i < total; i += gridDim.x * blockDim.x) {
    int c = i & 7;
    int t = (i >> 3) % 24;
    int n = i / 192;
    xp[i] = (c < 5) ? x[(size_t)n * 120 + c * 24 + t] : 0.0f;
  }
}

// W[CoutR][CinR][KK] -> B-fragment pair order:
//   wp[ ((kk*CI4 + c4)*2 + kup) * COUTP + co ][pair],  ci = c4*4 + kup*2 + pair
__global__ void pack_w(float* __restrict__ wp, const float* __restrict__ W,
                       int CoutR, int CinR, int KK, int CI4, int COUTP, int total) {
  for (int i = blockIdx.x * blockDim.x + threadIdx.x; i < total; i += gridDim.x * blockDim.x) {
    int pair = i & 1;
    int v2   = i >> 1;
    int co   = v2 % COUTP;
    int rest = v2 / COUTP;
    int kup  = rest & 1;
    int kc   = rest >> 1;
    int c4   = kc % CI4;
    int kk   = kc / CI4;
    int ci   = c4 * 4 + kup * 2 + pair;
    wp[i] = (co < CoutR && ci < CinR) ? W[((size_t)co * CinR + ci) * KK + kk] : 0.0f;
  }
}

// lin_w[24][2112] (f = ch*22 + j) -> time-major (f' = j*96 + ch) B-fragment pairs,
// padded to 32 output rows:  wp[ (c4*2 + kup)*32 + pcol ][pair]
__global__ void pack_lw(float* __restrict__ wp, const float* __restrict__ lw) {
  const int total = (LINF / 4) * 2 * 32 * 2;
  for (int i = blockIdx.x * blockDim.x + threadIdx.x; i < total; i += gridDim.x * blockDim.x) {
    int pair = i & 1;
    int v2   = i >> 1;
    int pcol = v2 & 31;
    int rest = v2 >> 5;
    int kup  = rest & 1;
    int c4   = rest >> 1;
    int fp   = c4 * 4 + kup * 2 + pair;   // time-major feature index
    int j    = fp / 96;
    int ch   = fp % 96;
    wp[i] = (pcol < PRED) ? lw[(size_t)pcol * LINF + ch * 22 + j] : 0.0f;
  }
}

// acc[n][j][co'] = co'<Cout ? b0+b1+b2+b_res[co'*Lout+j] : 0   (time-major)
__global__ void init_acc_k(float* __restrict__ acc, const float* __restrict__ b0,
                           const float* __restrict__ b1, const float* __restrict__ b2,
                           const float* __restrict__ brs, int total, int COUTP, int Cout, int Lout) {
  for (int i = blockIdx.x * blockDim.x + threadIdx.x; i < total; i += gridDim.x * blockDim.x) {
    int co = i % COUTP;
    int j  = (i / COUTP) % Lout;
    acc[i] = (co < Cout) ? b0[co] + b1[co] + b2[co] + brs[co * Lout + j] : 0.0f;
  }
}

// ----------------------------------------------------- WMMA conv (one term)
// Time-major: in[n][t][CST], acc[n][j][COUTP]; B pre-packed in fragment order.
// All operand fetches are branch-free (padding pre-applied); the causal t>=0
// guard is wave-uniform and skips zero-contribution WMMAs entirely.
template <int CIN, int CST, int Lin, int COUTP, int Lout, int KK, int PAD, bool LAST>
__global__ void __launch_bounds__(32) conv_term(float* __restrict__ acc,
                                                const float* __restrict__ in,
                                                const float* __restrict__ Wp) {
  constexpr int COT = COUTP / 16;
  constexpr int CI4 = CIN / 4;
  const int lane  = threadIdx.x;
  const int col   = lane & 15;
  const int kup   = lane >> 4;
  const int node  = blockIdx.x * 16 + col;          // A-fragment row
  const int cnode = blockIdx.x * 16 + (kup << 3);   // C/D row base (+p)
  const float* inrow = in + (size_t)node * (Lin * CST);
  const v2f* wp = (const v2f*)Wp;

#pragma unroll 1
  for (int j = 0; j < Lout; ++j) {
    v8f accv[COT];
#pragma unroll
    for (int ct = 0; ct < COT; ++ct)
#pragma unroll
      for (int p = 0; p < 8; ++p)
        accv[ct][p] = acc[(size_t)(cnode + p) * (Lout * COUTP) + j * COUTP + ct * 16 + col];

#pragma unroll
    for (int kk = 0; kk < KK; ++kk) {
      const int t = j + kk - PAD;
      if (PAD == 0 || t >= 0) {          // wave-uniform causal guard
#pragma unroll 1
        for (int c4 = 0; c4 < CI4; ++c4) {
          const v2f av = *(const v2f*)(inrow + t * CST + c4 * 4 + kup * 2);
          const int bbase = ((kk * CI4 + c4) * 2 + kup) * COUTP + col;
#pragma unroll
          for (int ct = 0; ct < COT; ++ct) {
            const v2f bv = wp[bbase + ct * 16];
            accv[ct] = __builtin_amdgcn_wmma_f32_16x16x4_f32(
                false, av, false, bv, (short)0, accv[ct], false, false);
          }
        }
      }
    }
#pragma unroll
    for (int ct = 0; ct < COT; ++ct)
#pragma unroll
      for (int p = 0; p < 8; ++p) {
        float v = accv[ct][p];
        if (LAST) v = fmaxf(v, 0.0f);    // biases + b_res folded in by init_acc_k
        acc[(size_t)(cnode + p) * (Lout * COUTP) + j * COUTP + ct * 16 + col] = v;
      }
  }
}

// ----------------------------------------------------- final linear (WMMA)
// out[n,p] = sum_f h[n,f'] * lwp + lb[p];  h is time-major [N][2112]
__global__ void __launch_bounds__(32) linear24(const float* __restrict__ h,
                                               const float* __restrict__ lwp,
                                               const float* __restrict__ lb,
                                               float* __restrict__ out) {
  const int lane  = threadIdx.x;
  const int col   = lane & 15;
  const int kup   = lane >> 4;
  const int node  = blockIdx.x * 16 + col;
  const int cnode = blockIdx.x * 16 + (kup << 3);
  const int p1    = 16 + col;
  const float* hrow = h + (size_t)node * LINF;
  const v2f* wp = (const v2f*)lwp;
  v8f a0 = {};
  v8f a1 = {};
#pragma unroll 4
  for (int c4 = 0; c4 < LINF / 4; ++c4) {
    const v2f av  = *(const v2f*)(hrow + c4 * 4 + kup * 2);
    const int bb  = (c4 * 2 + kup) * 32 + col;
    const v2f b0v = wp[bb];
    const v2f b1v = wp[bb + 16];
    a0 = __builtin_amdgcn_wmma_f32_16x16x4_f32(false, av, false, b0v, (short)0, a0, false, false);
    a1 = __builtin_amdgcn_wmma_f32_16x16x4_f32(false, av, false, b1v, (short)0, a1, false, false);
  }
#pragma unroll
  for (int p = 0; p < 8; ++p) {
    out[(size_t)(cnode + p) * PRED + col] = a0[p] + lb[col];
    if (p1 < PRED) out[(size_t)(cnode + p) * PRED + p1] = a1[p] + lb[p1];
  }
}

// ------------------------------------------------------------- layer driver
template <int CIN, int CST, int Lin, int CoutR, int CinR, int COUTP, int Lout, int KK, int PAD>
static void run_layer(const float* h, float* acc, float* T1, float* T2,
                      const float* W0, const float* b0,
                      const float* W1, const float* b1,
                      const float* W2, const float* b2,
                      const float* brs,
                      const int* src, const int* dst, const float* wgt,
                      float* wp0, float* wp1, float* wp2, hipStream_t stream) {
  constexpr int F  = Lin * CST;
  constexpr int OT = Lout * COUTP;
  constexpr int CI4 = CIN / 4;
  constexpr int WPT = KK * CI4 * 2 * COUTP * 2;
  const size_t nF = (size_t)N_NODES * F;
  // pack the three term weights into B-fragment order (zero padded)
  pack_w<<<(WPT + 255) / 256, 256, 0, stream>>>(wp0, W0, CoutR, CinR, KK, CI4, COUTP, WPT);
  pack_w<<<(WPT + 255) / 256, 256, 0, stream>>>(wp1, W1, CoutR, CinR, KK, CI4, COUTP, WPT);
  pack_w<<<(WPT + 255) / 256, 256, 0, stream>>>(wp2, W2, CoutR, CinR, KK, CI4, COUTP, WPT);
  // T1 = Lx(h) = h + S(h)
  axpby<<<2048, 256, 0, stream>>>(T1, h, nullptr, 1.0f, 0.0f, nF);
  scatter_add<<<2500, 256, 0, stream>>>(T1, h, src, dst, wgt, 1.0f, N_EDGES, F);
  // T2 = 2*Lx(T1) - h = (2*T1 - h) + 2*S(T1)
  axpby<<<2048, 256, 0, stream>>>(T2, T1, h, 2.0f, -1.0f, nF);
  scatter_add<<<2500, 256, 0, stream>>>(T2, T1, src, dst, wgt, 2.0f, N_EDGES, F);
  // acc = biases (+ b_res), padded channels = 0; three GEMM terms; last fuses ReLU
  init_acc_k<<<2048, 256, 0, stream>>>(acc, b0, b1, b2, brs, N_NODES * OT, COUTP, CoutR, Lout);
  conv_term<CIN, CST, Lin, COUTP, Lout, KK, PAD, false><<<N_NODES / 16, 32, 0, stream>>>(acc, h,  wp0);
  conv_term<CIN, CST, Lin, COUTP, Lout, KK, PAD, false><<<N_NODES / 16, 32, 0, stream>>>(acc, T1, wp1);
  conv_term<CIN, CST, Lin, COUTP, Lout, KK, PAD, true ><<<N_NODES / 16, 32, 0, stream>>>(acc, T2, wp2);
}

extern "C" void kernel_launch(void* const* d_in, const int* in_sizes, int n_in,
                              void* d_out, int out_size, void* d_ws, size_t ws_size,
                              hipStream_t stream) {
  static const int cfgs[6][6] = {{5, 24, 24, 32, 5, 12}, {24, 32, 32, 30, 5, 2},
                                 {32, 30, 64, 28, 5, 2}, {64, 28, 64, 26, 3, 0},
                                 {64, 26, 96, 24, 3, 0}, {96, 24, 96, 22, 3, 0}};
  // ---- identify inputs by element count (robust to pytree flatten order) ----
  const float* x = nullptr; const int* ei = nullptr;
  const float* lw = nullptr; const float* lb = nullptr;
  const float* W[6][3] = {}; const float* B[6][3] = {}; const float* BR[6] = {};
  char claimed[256] = {0};
  const int nn = (n_in < 256) ? n_in : 256;
  for (int i = 0; i < nn; ++i) {
    if (!x  && in_sizes[i] == N_NODES * 120) { x  = (const float*)d_in[i]; claimed[i] = 1; continue; }
    if (!ei && in_sizes[i] == 2 * N_EDGES)   { ei = (const int*)d_in[i];   claimed[i] = 1; continue; }
    if (!lw && in_sizes[i] == PRED * LINF)   { lw = (const float*)d_in[i]; claimed[i] = 1; continue; }
  }
  for (int l = 0; l < 6; ++l) {   // W sizes unique per layer; bias follows its W (tuple adjacency)
    const int wsz = cfgs[l][2] * cfgs[l][0] * cfgs[l][4];
    int t = 0;
    for (int i = 0; i < nn && t < 3; ++i) {
      if (!claimed[i] && in_sizes[i] == wsz) {
        claimed[i] = 1; W[l][t] = (const float*)d_in[i];
        if (i + 1 < nn && !claimed[i + 1] && in_sizes[i + 1] == cfgs[l][2]) {
          claimed[i + 1] = 1; B[l][t] = (const float*)d_in[i + 1];
        }
        ++t;
      }
    }
  }
  for (int l = 0; l < 6; ++l)     // fallback if biases weren't adjacent
    for (int t = 0; t < 3; ++t)
      if (W[l][t] && !B[l][t])
        for (int i = 0; i < nn; ++i)
          if (!claimed[i] && in_sizes[i] == cfgs[l][2]) { claimed[i] = 1; B[l][t] = (const float*)d_in[i]; break; }
  for (int l = 0; l < 6; ++l) {   // b_res sizes unique
    const int sz = cfgs[l][2] * cfgs[l][3];
    for (int i = 0; i < nn; ++i)
      if (!claimed[i] && in_sizes[i] == sz) { claimed[i] = 1; BR[l] = (const float*)d_in[i]; break; }
  }
  for (int i = 0; i < nn; ++i)
    if (!claimed[i] && in_sizes[i] == PRED) { claimed[i] = 1; lb = (const float*)d_in[i]; break; }

  if (!x || !ei || !lw || !lb) return;
  for (int l = 0; l < 6; ++l) {
    if (!BR[l]) return;
    for (int t = 0; t < 3; ++t) if (!W[l][t] || !B[l][t]) return;
  }

  // ---- workspace layout ----
  const size_t NB    = (size_t)N_NODES * MAXF;
  const size_t XPSZ  = (size_t)N_NODES * 24 * 8;        // padded x
  const size_t WPSLOT = 32768;                          // >= largest packed term (27648)
  const size_t LWPSZ = (size_t)(LINF / 4) * 2 * 32 * 2; // 67584
  const size_t need  = (4 * NB + N_NODES + N_EDGES + XPSZ + 18 * WPSLOT + LWPSZ) * sizeof(float);
  if (ws_size < need) return;
  float* bufA = (float*)d_ws;
  float* bufB = bufA + NB;
  float* T1   = bufB + NB;
  float* T2   = T1 + NB;
  float* deg  = T2 + NB;
  float* wgt  = deg + N_NODES;
  float* xp   = wgt + N_EDGES;
  float* wpk  = xp + XPSZ;                              // 18 slots
  float* lwp  = wpk + 18 * WPSLOT;
  const int* src = ei;
  const int* dst = ei + N_EDGES;
  float* wp[6][3];
  for (int l = 0; l < 6; ++l)
    for (int t = 0; t < 3; ++t) wp[l][t] = wpk + (size_t)(l * 3 + t) * WPSLOT;

  // ---- one-time prep: edge weights, padded x, packed linear weights ----
  zero_f<<<64, 256, 0, stream>>>(deg, N_NODES);
  deg_acc<<<(N_EDGES + 255) / 256, 256, 0, stream>>>(deg, src, N_EDGES);
  dinv_k<<<(N_NODES + 255) / 256, 256, 0, stream>>>(deg, N_NODES);
  edgew<<<(N_EDGES + 255) / 256, 256, 0, stream>>>(wgt, deg, src, dst, N_EDGES);
  pack_x<<<2048, 256, 0, stream>>>(xp, x);
  pack_lw<<<264, 256, 0, stream>>>(lwp, lw);

  // ---- 6 ChebConv layers, time-major padded layouts ----
  // <CIN, CST, Lin, CoutR, CinR, COUTP, Lout, KK, PAD>
  run_layer< 8,  8, 24, 24,  5, 32, 32, 5, 12>(xp,   bufA, T1, T2, W[0][0], B[0][0], W[0][1], B[0][1], W[0][2], B[0][2], BR[0], src, dst, wgt, wp[0][0], wp[0][1], wp[0][2], stream);
  run_layer<24, 32, 32, 32, 24, 32, 30, 5,  2>(bufA, bufB, T1, T2, W[1][0], B[1][0], W[1][1], B[1][1], W[1][2], B[1][2], BR[1], src, dst, wgt, wp[1][0], wp[1][1], wp[1][2], stream);
  run_layer<32, 32, 30, 64, 32, 64, 28, 5,  2>(bufB, bufA, T1, T2, W[2][0], B[2][0], W[2][1], B[2][1], W[2][2], B[2][2], BR[2], src, dst, wgt, wp[2][0], wp[2][1], wp[2][2], stream);
  run_layer<64, 64, 28, 64, 64, 64, 26, 3,  0>(bufA, bufB, T1, T2, W[3][0], B[3][0], W[3][1], B[3][1], W[3][2], B[3][2], BR[3], src, dst, wgt, wp[3][0], wp[3][1], wp[3][2], stream);
  run_layer<64, 64, 26, 96, 64, 96, 24, 3,  0>(bufB, bufA, T1, T2, W[4][0], B[4][0], W[4][1], B[4][1], W[4][2], B[4][2], BR[4], src, dst, wgt, wp[4][0], wp[4][1], wp[4][2], stream);
  run_layer<96, 96, 24, 96, 96, 96, 22, 3,  0>(bufA, bufB, T1, T2, W[5][0], B[5][0], W[5][1], B[5][1], W[5][2], B[5][2], BR[5], src, dst, wgt, wp[5][0], wp[5][1], wp[5][2], stream);

  // ---- final linear: (10000 x 2112) @ (2112 x 24) + bias ----
  linear24<<<N_NODES / 16, 32, 0, stream>>>(bufB, lwp, lb, (float*)d_out);
}